// TinyRGATEncoder_30614526885987
// MI455X (gfx1250) — compile-verified
//
#include <hip/hip_runtime.h>
#include <hip/hip_bf16.h>

typedef __attribute__((ext_vector_type(2))) float v2f;
typedef __attribute__((ext_vector_type(8))) float v8f;

#define H_DIM 64
#define ENC_NEG_INF 0x007FFFFFu

// ---- order-preserving float<->uint encoding for atomic segment-max ----
__device__ __forceinline__ unsigned enc_f32(float f) {
  unsigned u = __float_as_uint(f);
  return (u & 0x80000000u) ? ~u : (u | 0x80000000u);
}
__device__ __forceinline__ float dec_f32(unsigned k) {
  unsigned u = (k & 0x80000000u) ? (k & 0x7FFFFFFFu) : ~k;
  return __uint_as_float(u);
}

// ---- fp32 WMMA GEMM:  C[N,64] = act(A[N,K] @ W[K,64] (+ bias)) ----
// one wave = 16 rows x 64 cols; V_WMMA_F32_16X16X4_F32, exact fp32 math.
// A-frag (16x4 f32): lane m=l&15, hi=l>>4 -> contiguous float2 at col k+2*hi.
// B-frag (4x16 f32): lane n=l&15      -> W[k+2*hi][n], W[k+2*hi+1][n].
// C/D  (16x16 f32): vgpr r -> row r+8*hi, col n (+16*tile).
template <int K, bool RELU>
__global__ void k_gemm(const float* __restrict__ A, const float* __restrict__ W,
                       const float* __restrict__ bvec, float* __restrict__ C, int N) {
  __shared__ float sW[K * H_DIM];
  for (int i = threadIdx.x; i < K * H_DIM; i += blockDim.x) sW[i] = W[i];
  __syncthreads();

  const int wave = threadIdx.x >> 5;
  const int lane = threadIdx.x & 31;
  const int row0 = (blockIdx.x * 8 + wave) * 16;
  if (row0 >= N) return;  // wave-uniform: EXEC stays all-ones for WMMA

  const int m = lane & 15;
  const int hi = lane >> 4;
  v8f acc[4] = {v8f{}, v8f{}, v8f{}, v8f{}};
  const float* __restrict__ arow = A + (size_t)(row0 + m) * K;

  for (int k = 0; k < K; k += 4) {
    const int kb = k + 2 * hi;
    v2f a;
    a.x = arow[kb];
    a.y = arow[kb + 1];
#pragma unroll
    for (int t = 0; t < 4; ++t) {
      const int n = m + 16 * t;
      v2f b;
      b.x = sW[kb * H_DIM + n];
      b.y = sW[(kb + 1) * H_DIM + n];
      acc[t] = __builtin_amdgcn_wmma_f32_16x16x4_f32(
          false, a, false, b, (short)0, acc[t], false, false);
    }
  }

#pragma unroll
  for (int t = 0; t < 4; ++t) {
    const int n = m + 16 * t;
    const float bb = RELU ? bvec[n] : 0.0f;
#pragma unroll
    for (int r = 0; r < 8; ++r) {
      float v = acc[t][r] + bb;
      if (RELU) v = v > 0.0f ? v : 0.0f;
      C[(size_t)(row0 + r + 8 * hi) * H_DIM + n] = v;
    }
  }
}

// ---- per-relation attention scalar: erel[r] = (rel_emb[r] @ W_relproj) @ a_rel ----
__global__ void k_rel(const float* __restrict__ re, const float* __restrict__ Wp,
                      const float* __restrict__ a_rel, float* __restrict__ erel) {
  int r = threadIdx.x;
  if (r < 8) {
    float acc = 0.0f;
    for (int j = 0; j < H_DIM; ++j) {
      float pj = 0.0f;
      for (int k = 0; k < 16; ++k) pj += re[r * 16 + k] * Wp[k * H_DIM + j];
      acc += pj * a_rel[j];
    }
    erel[r] = acc;
  }
}

// ---- per-node attention scalars: ed = hm@a_dst, es = hm@a_src (wave per node) ----
__global__ void k_node_scores(const float* __restrict__ hm,
                              const float* __restrict__ a_dst,
                              const float* __restrict__ a_src,
                              float* __restrict__ ed, float* __restrict__ es, int N) {
  const int wave = threadIdx.x >> 5, lane = threadIdx.x & 31;
  const int n = blockIdx.x * (blockDim.x >> 5) + wave;
  if (n >= N) return;
  float2 v = ((const float2*)(hm + (size_t)n * H_DIM))[lane];
  float2 ad = ((const float2*)a_dst)[lane];
  float2 as = ((const float2*)a_src)[lane];
  float pd = v.x * ad.x + v.y * ad.y;
  float ps = v.x * as.x + v.y * as.y;
#pragma unroll
  for (int o = 16; o >= 1; o >>= 1) {
    pd += __shfl_xor(pd, o, 32);
    ps += __shfl_xor(ps, o, 32);
  }
  if (lane == 0) { ed[n] = pd; es[n] = ps; }
}

// ---- per-layer init: out=0, m=-inf(key), s=0 ----
__global__ void k_init(unsigned* __restrict__ mkey, float* __restrict__ s,
                       float* __restrict__ out, int N) {
  const int i = blockIdx.x * blockDim.x + threadIdx.x;
  if (i < N * H_DIM) out[i] = 0.0f;
  if (i < N) { mkey[i] = ENC_NEG_INF; s[i] = 0.0f; }
}

// ---- edge pass A: logits + segment-max ----
__global__ void k_passA(const int* __restrict__ src, const int* __restrict__ dst,
                        const int* __restrict__ et, const float* __restrict__ conf,
                        const float* __restrict__ es, const float* __restrict__ ed,
                        const float* __restrict__ erel, float* __restrict__ escr,
                        unsigned* __restrict__ mkey, int E) {
  const int i = blockIdx.x * blockDim.x + threadIdx.x;
  if (i >= E) return;
  const int dN = dst[i];
  int t = et[i];
  t = t < 0 ? 0 : (t > 7 ? 7 : t);
  float e = ed[dN] + es[src[i]] + erel[t];
  e = e > 0.0f ? e : 0.2f * e;                 // leaky_relu(0.2)
  e += 0.1f * __logf(fmaxf(conf[i], 1e-6f));   // CONF_LOG_WEIGHT
  escr[i] = e;
  atomicMax(&mkey[dN], enc_f32(e));
}

// ---- edge pass B: exp(e - m[dst]) + segment-sum ----
__global__ void k_passB(const int* __restrict__ dst, const float* __restrict__ escr,
                        const unsigned* __restrict__ mkey, float* __restrict__ exscr,
                        float* __restrict__ s, int E) {
  const int i = blockIdx.x * blockDim.x + threadIdx.x;
  if (i >= E) return;
  const int dN = dst[i];
  const float ex = __expf(escr[i] - dec_f32(mkey[dN]));
  exscr[i] = ex;
  atomicAdd(&s[dN], ex);
}

// ---- edge pass C: out[dst] += alpha * hm[src]   (wave per edge, float2/lane) ----
__global__ void k_passC(const int* __restrict__ src, const int* __restrict__ dst,
                        const float* __restrict__ exscr, const float* __restrict__ s,
                        const float* __restrict__ hm, float* __restrict__ out,
                        long long E) {
  const int wave = threadIdx.x >> 5, lane = threadIdx.x & 31;
  const long long i = (long long)blockIdx.x * (blockDim.x >> 5) + wave;
  if (i >= E) return;
  const int sN = src[i], dN = dst[i];
  const float alpha = exscr[i] / s[dN];  // broadcast loads (same addr per wave)
  const float2 hv = ((const float2*)(hm + (size_t)sN * H_DIM))[lane];
  float* op = out + (size_t)dN * H_DIM + lane * 2;
  atomicAdd(op, alpha * hv.x);
  atomicAdd(op + 1, alpha * hv.y);
}

// ---- node update: h = layernorm(h + relu(out + bias)) (wave per node) ----
__global__ void k_update(float* __restrict__ h, const float* __restrict__ out,
                         const float* __restrict__ bias, const float* __restrict__ g,
                         const float* __restrict__ b, int N) {
  const int wave = threadIdx.x >> 5, lane = threadIdx.x & 31;
  const int n = blockIdx.x * (blockDim.x >> 5) + wave;
  if (n >= N) return;
  const float2 hv = ((const float2*)(h + (size_t)n * H_DIM))[lane];
  const float2 ov = ((const float2*)(out + (size_t)n * H_DIM))[lane];
  const float2 bb = ((const float2*)bias)[lane];
  float vx = ov.x + bb.x; vx = vx > 0.0f ? vx : 0.0f;
  float vy = ov.y + bb.y; vy = vy > 0.0f ? vy : 0.0f;
  const float tx = hv.x + vx, ty = hv.y + vy;
  float sum = tx + ty, sq = tx * tx + ty * ty;
#pragma unroll
  for (int o = 16; o >= 1; o >>= 1) {
    sum += __shfl_xor(sum, o, 32);
    sq += __shfl_xor(sq, o, 32);
  }
  const float mu = sum * (1.0f / 64.0f);
  const float var = sq * (1.0f / 64.0f) - mu * mu;
  const float inv = __frsqrt_rn(var + 1e-5f);
  const float2 gg = ((const float2*)g)[lane];
  const float2 b2 = ((const float2*)b)[lane];
  float2 o2;
  o2.x = (tx - mu) * inv * gg.x + b2.x;
  o2.y = (ty - mu) * inv * gg.y + b2.y;
  ((float2*)(h + (size_t)n * H_DIM))[lane] = o2;
}

extern "C" void kernel_launch(void* const* d_in, const int* in_sizes, int n_in,
                              void* d_out, int out_size, void* d_ws, size_t ws_size,
                              hipStream_t stream) {
  const float* x         = (const float*)d_in[0];
  const float* W_in      = (const float*)d_in[1];
  const float* b_in      = (const float*)d_in[2];
  const float* W_msg     = (const float*)d_in[3];
  const float* rel_emb   = (const float*)d_in[4];
  const float* W_relproj = (const float*)d_in[5];
  const float* att       = (const float*)d_in[6];
  const float* bias      = (const float*)d_in[7];
  const float* ln_g      = (const float*)d_in[8];
  const float* ln_b      = (const float*)d_in[9];
  const float* edge_attr = (const float*)d_in[10];
  const int*   edge_index= (const int*)d_in[11];
  const int*   edge_type = (const int*)d_in[12];

  const int IN = 128;
  const int N = in_sizes[0] / IN;
  const int E = in_sizes[12];
  const int L = in_sizes[3] / (H_DIM * H_DIM);
  const int* src = edge_index;
  const int* dst = edge_index + E;

  float* h = (float*)d_out;  // h lives in d_out, updated in place per layer

  // workspace carve-up
  char* ws = (char*)d_ws;
  float*    hm    = (float*)ws;    ws += (size_t)N * H_DIM * sizeof(float);
  float*    out   = (float*)ws;    ws += (size_t)N * H_DIM * sizeof(float);
  float*    es    = (float*)ws;    ws += (size_t)N * sizeof(float);
  float*    ed    = (float*)ws;    ws += (size_t)N * sizeof(float);
  unsigned* mkey  = (unsigned*)ws; ws += (size_t)N * sizeof(unsigned);
  float*    s     = (float*)ws;    ws += (size_t)N * sizeof(float);
  float*    escr  = (float*)ws;    ws += (size_t)E * sizeof(float);
  float*    exscr = (float*)ws;    ws += (size_t)E * sizeof(float);
  float*    erel  = (float*)ws;    ws += 8 * sizeof(float);

  const int tiles = (N + 15) / 16;
  const int gemm_blocks = (tiles + 7) / 8;       // 8 waves/block, 16 rows/wave
  const int node_wave_blocks = (N + 7) / 8;      // 8 nodes/block (wave per node)
  const int init_blocks = (N * H_DIM + 255) / 256;
  const int edge_blocks = (E + 255) / 256;
  const int edge_wave_blocks = (E + 7) / 8;      // wave per edge

  // h = relu(x @ W_in + b_in)   [fp32 WMMA]
  k_gemm<128, true><<<gemm_blocks, 256, 0, stream>>>(x, W_in, b_in, h, N);

  for (int l = 0; l < L; ++l) {
    const float* a_dst = att + (size_t)l * 3 * H_DIM;
    const float* a_src = a_dst + H_DIM;
    const float* a_rel = a_dst + 2 * H_DIM;

    k_rel<<<1, 64, 0, stream>>>(rel_emb + (size_t)l * 8 * 16,
                                W_relproj + (size_t)l * 16 * H_DIM, a_rel, erel);
    // hm = h @ W_msg[l]   [fp32 WMMA]
    k_gemm<64, false><<<gemm_blocks, 256, 0, stream>>>(
        h, W_msg + (size_t)l * H_DIM * H_DIM, nullptr, hm, N);
    k_node_scores<<<node_wave_blocks, 256, 0, stream>>>(hm, a_dst, a_src, ed, es, N);
    k_init<<<init_blocks, 256, 0, stream>>>(mkey, s, out, N);
    k_passA<<<edge_blocks, 256, 0, stream>>>(src, dst, edge_type, edge_attr,
                                             es, ed, erel, escr, mkey, E);
    k_passB<<<edge_blocks, 256, 0, stream>>>(dst, escr, mkey, exscr, s, E);
    k_passC<<<edge_wave_blocks, 256, 0, stream>>>(src, dst, exscr, s, hm, out,
                                                  (long long)E);
    k_update<<<node_wave_blocks, 256, 0, stream>>>(
        h, out, bias + (size_t)l * H_DIM, ln_g + (size_t)l * H_DIM,
        ln_b + (size_t)l * H_DIM, N);
  }
}